// HGCN_41695542509880
// MI455X (gfx1250) — compile-verified
//
#include <hip/hip_runtime.h>

typedef __attribute__((ext_vector_type(2))) float v2f;
typedef __attribute__((ext_vector_type(8))) float v8f;

#define HD 128  // feature/hidden dim (D == H == 128 in this model)

// ---------------------------------------------------------------- utilities
__global__ __launch_bounds__(256) void zero_f32(float* __restrict__ p, long n) {
  long i = (long)blockIdx.x * blockDim.x + threadIdx.x;
  const long stride = (long)gridDim.x * blockDim.x;
  for (; i < n; i += stride) p[i] = 0.0f;
}

__global__ __launch_bounds__(256) void degree_accum(const int* __restrict__ src,
                                                    const int* __restrict__ dst,
                                                    float* __restrict__ deg_out,
                                                    float* __restrict__ deg_in,
                                                    int nedges) {
  int i = blockIdx.x * blockDim.x + threadIdx.x;
  const int stride = gridDim.x * blockDim.x;
  for (; i < nedges; i += stride) {
    atomicAdd(&deg_out[src[i]], 1.0f);
    atomicAdd(&deg_in[dst[i]], 1.0f);
  }
}

__global__ __launch_bounds__(256) void inv_sqrt_deg(float* __restrict__ p, int n) {
  int i = blockIdx.x * blockDim.x + threadIdx.x;
  const int stride = gridDim.x * blockDim.x;
  for (; i < n; i += stride) p[i] = rsqrtf(fmaxf(p[i], 1.0f));
}

// ---------------------------------------------------------------- GEMM (WMMA f32)
// Y[nrows,128] = X[nrows,128] @ W[128,128]
// Block = 128 threads (4 waves, wave32). W cached in LDS (64 KB << 320 KB/WGP).
// Each wave owns a 32-column strip (two 16x16 WMMA tiles). Each block
// iteration covers a 32-row tile (two stacked A fragments), so every set of
// B fragments from LDS feeds FOUR v_wmma_f32_16x16x4_f32 ops.
__global__ __launch_bounds__(128) void gemm_n128(const float* __restrict__ X,
                                                 const float* __restrict__ W,
                                                 float* __restrict__ Y,
                                                 int nrows) {
  __shared__ float Ws[HD * HD];
  for (int i = threadIdx.x * 4; i < HD * HD; i += blockDim.x * 4)
    *(float4*)(Ws + i) = *(const float4*)(W + i);
  __syncthreads();

  const int lane = threadIdx.x & 31;
  const int wave = threadIdx.x >> 5;
  const int lhi = lane >> 4;   // 0 -> lanes 0-15, 1 -> lanes 16-31
  const int llo = lane & 15;
  const int n0 = wave * 32;    // this wave's column strip
  const int ntiles = (nrows + 31) >> 5;

  for (int t = blockIdx.x; t < ntiles; t += gridDim.x) {
    const int row0 = t << 5;
    // A-fragment source rows for the two stacked 16-row tiles
    int arA = row0 + llo;
    int arB = row0 + 16 + llo;
    if (arA >= nrows) arA = nrows - 1;   // clamp (tail tile only; stores guarded)
    if (arB >= nrows) arB = nrows - 1;
    const float* xrA = X + (size_t)arA * HD;
    const float* xrB = X + (size_t)arB * HD;

    v8f accA0 = {};
    v8f accA1 = {};
    v8f accB0 = {};
    v8f accB1 = {};
    #pragma unroll 4
    for (int k0 = 0; k0 < HD; k0 += 4) {
      // A 16x4 f32 fragments: lanes 0-15 hold K=k0,k0+1; lanes 16-31 K=k0+2,k0+3
      v2f aA, aB;
      aA.x = xrA[k0 + 2 * lhi];
      aA.y = xrA[k0 + 2 * lhi + 1];
      aB.x = xrB[k0 + 2 * lhi];
      aB.y = xrB[k0 + 2 * lhi + 1];
      // B 4x16 fragments: VGPR0 rows K=k0 (lo lanes)/k0+2 (hi lanes), VGPR1 K=k0+1/k0+3
      const int kA = (k0 + 2 * lhi) * HD;
      const int kB = (k0 + 2 * lhi + 1) * HD;
      v2f b0, b1;
      b0.x = Ws[kA + n0 + llo];
      b0.y = Ws[kB + n0 + llo];
      b1.x = Ws[kA + n0 + 16 + llo];
      b1.y = Ws[kB + n0 + 16 + llo];
      accA0 = __builtin_amdgcn_wmma_f32_16x16x4_f32(false, aA, false, b0,
                                                    (short)0, accA0, false, false);
      accA1 = __builtin_amdgcn_wmma_f32_16x16x4_f32(false, aA, false, b1,
                                                    (short)0, accA1, false, false);
      accB0 = __builtin_amdgcn_wmma_f32_16x16x4_f32(false, aB, false, b0,
                                                    (short)0, accB0, false, false);
      accB1 = __builtin_amdgcn_wmma_f32_16x16x4_f32(false, aB, false, b1,
                                                    (short)0, accB1, false, false);
    }
    // D layout: VGPR r -> (M=mbase+r, N=llo) lanes 0-15, (M=mbase+8+r) lanes 16-31
    if (row0 + 32 <= nrows) {
      // fast path: tile fully in range, branch-free coalesced stores
      float* y0 = Y + (size_t)(row0 + 8 * lhi) * HD + n0 + llo;
      float* y1 = Y + (size_t)(row0 + 16 + 8 * lhi) * HD + n0 + llo;
      #pragma unroll
      for (int r = 0; r < 8; ++r) {
        y0[(size_t)r * HD]      = accA0[r];
        y0[(size_t)r * HD + 16] = accA1[r];
        y1[(size_t)r * HD]      = accB0[r];
        y1[(size_t)r * HD + 16] = accB1[r];
      }
    } else {
      #pragma unroll
      for (int r = 0; r < 8; ++r) {
        const int mA = row0 + r + 8 * lhi;
        const int mB = row0 + 16 + r + 8 * lhi;
        if (mA < nrows) {
          Y[(size_t)mA * HD + n0 + llo]      = accA0[r];
          Y[(size_t)mA * HD + n0 + 16 + llo] = accA1[r];
        }
        if (mB < nrows) {
          Y[(size_t)mB * HD + n0 + llo]      = accB0[r];
          Y[(size_t)mB * HD + n0 + 16 + llo] = accB1[r];
        }
      }
    }
  }
}

// ---------------------------------------------------------------- edge scatter
// One wave32 per edge: lane moves a float4 -> 512B coalesced gather and 128
// contiguous f32 atomic adds per edge. Both relations accumulate into the
// same `agg` buffer (mean/bias folded in later).
__global__ __launch_bounds__(256) void scatter_edges(const float* __restrict__ Hp,
                                                     const int* __restrict__ src,
                                                     const int* __restrict__ dst,
                                                     const float* __restrict__ inv_out,
                                                     const float* __restrict__ inv_in,
                                                     float* __restrict__ agg,
                                                     int nedges) {
  const int lane = threadIdx.x & 31;
  int e = blockIdx.x * (blockDim.x >> 5) + (threadIdx.x >> 5);
  const int stride = gridDim.x * (blockDim.x >> 5);
  for (; e < nedges; e += stride) {
    const int s = src[e];
    const int d = dst[e];
    // prefetch next edge's source row to hide gather latency
    const int en = e + stride;
    if (en < nedges)
      __builtin_prefetch(Hp + (size_t)src[en] * HD + lane * 4, 0, 0);
    const float c = inv_out[s] * inv_in[d];
    const float4 v = *((const float4*)(Hp + (size_t)s * HD) + lane);
    float* ap = agg + (size_t)d * HD + lane * 4;
    atomicAdd(ap + 0, v.x * c);
    atomicAdd(ap + 1, v.y * c);
    atomicAdd(ap + 2, v.z * c);
    atomicAdd(ap + 3, v.w * c);
  }
}

// ---------------------------------------------------------------- combine
// h = relu(0.5*(agg_ppi + b_p + agg_sim + b_s)), agg already = agg_ppi+agg_sim
__global__ __launch_bounds__(256) void combine_relu(const float* __restrict__ agg,
                                                    const float* __restrict__ bp,
                                                    const float* __restrict__ bs,
                                                    float* __restrict__ out,
                                                    long n4) {
  long i = (long)blockIdx.x * blockDim.x + threadIdx.x;
  const long stride = (long)gridDim.x * blockDim.x;
  for (; i < n4; i += stride) {
    const int col = (int)((i * 4) & (HD - 1));
    const float4 av  = ((const float4*)agg)[i];
    const float4 bpv = *(const float4*)(bp + col);
    const float4 bsv = *(const float4*)(bs + col);
    float4 o;
    o.x = fmaxf(0.0f, 0.5f * (av.x + bpv.x + bsv.x));
    o.y = fmaxf(0.0f, 0.5f * (av.y + bpv.y + bsv.y));
    o.z = fmaxf(0.0f, 0.5f * (av.z + bpv.z + bsv.z));
    o.w = fmaxf(0.0f, 0.5f * (av.w + bpv.w + bsv.w));
    ((float4*)out)[i] = o;
  }
}

// ---------------------------------------------------------------- classifier
// logits[p] = [z[s], z[d]] @ Wc + bc ; one wave32 per pair, Wc (2KB) in LDS.
__global__ __launch_bounds__(256) void classifier(const float* __restrict__ z,
                                                  const int* __restrict__ ps,
                                                  const int* __restrict__ pd,
                                                  const float* __restrict__ Wc,
                                                  const float* __restrict__ bc,
                                                  float* __restrict__ out,
                                                  int npairs) {
  __shared__ float wcl[2 * HD * 2];
  __shared__ float bcl[2];
  for (int i = threadIdx.x; i < 2 * HD * 2; i += blockDim.x) wcl[i] = Wc[i];
  if (threadIdx.x < 2) bcl[threadIdx.x] = bc[threadIdx.x];
  __syncthreads();

  const int lane = threadIdx.x & 31;
  int p = blockIdx.x * (blockDim.x >> 5) + (threadIdx.x >> 5);
  const int stride = gridDim.x * (blockDim.x >> 5);
  for (; p < npairs; p += stride) {
    const int s = ps[p];
    const int d = pd[p];
    const float4 zs = *((const float4*)(z + (size_t)s * HD) + lane);
    const float4 zd = *((const float4*)(z + (size_t)d * HD) + lane);
    const int j = lane * 4;
    float a0 = zs.x * wcl[(j + 0) * 2] + zs.y * wcl[(j + 1) * 2] +
               zs.z * wcl[(j + 2) * 2] + zs.w * wcl[(j + 3) * 2] +
               zd.x * wcl[(HD + j + 0) * 2] + zd.y * wcl[(HD + j + 1) * 2] +
               zd.z * wcl[(HD + j + 2) * 2] + zd.w * wcl[(HD + j + 3) * 2];
    float a1 = zs.x * wcl[(j + 0) * 2 + 1] + zs.y * wcl[(j + 1) * 2 + 1] +
               zs.z * wcl[(j + 2) * 2 + 1] + zs.w * wcl[(j + 3) * 2 + 1] +
               zd.x * wcl[(HD + j + 0) * 2 + 1] + zd.y * wcl[(HD + j + 1) * 2 + 1] +
               zd.z * wcl[(HD + j + 2) * 2 + 1] + zd.w * wcl[(HD + j + 3) * 2 + 1];
    #pragma unroll
    for (int off = 16; off > 0; off >>= 1) {
      a0 += __shfl_xor(a0, off, 32);
      a1 += __shfl_xor(a1, off, 32);
    }
    if (lane == 0) {
      out[(size_t)p * 2 + 0] = a0 + bcl[0];
      out[(size_t)p * 2 + 1] = a1 + bcl[1];
    }
  }
}

// ---------------------------------------------------------------- launch
static inline int grid_cap(long work, int per_block, int cap) {
  long g = (work + per_block - 1) / per_block;
  if (g > cap) g = cap;
  if (g < 1) g = 1;
  return (int)g;
}

extern "C" void kernel_launch(void* const* d_in, const int* in_sizes, int n_in,
                              void* d_out, int out_size, void* d_ws, size_t ws_size,
                              hipStream_t stream) {
  const int N  = in_sizes[0] / HD;
  const int E1 = in_sizes[1];   // ppi edges
  const int E2 = in_sizes[3];   // sim edges
  const int P  = in_sizes[5];   // query pairs

  const float* feat    = (const float*)d_in[0];
  const int* src_ppi   = (const int*)d_in[1];
  const int* dst_ppi   = (const int*)d_in[2];
  const int* src_sim   = (const int*)d_in[3];
  const int* dst_sim   = (const int*)d_in[4];
  const int* pair_src  = (const int*)d_in[5];
  const int* pair_dst  = (const int*)d_in[6];
  const float* W0_ppi  = (const float*)d_in[7];
  const float* b0_ppi  = (const float*)d_in[8];
  const float* W0_sim  = (const float*)d_in[9];
  const float* b0_sim  = (const float*)d_in[10];
  const float* W1_ppi  = (const float*)d_in[11];
  const float* b1_ppi  = (const float*)d_in[12];
  const float* W1_sim  = (const float*)d_in[13];
  const float* b1_sim  = (const float*)d_in[14];
  const float* Wc      = (const float*)d_in[15];
  const float* bc      = (const float*)d_in[16];

  float* zout   = (float*)d_out;             // [N, 128] final embeddings
  float* logits = zout + (size_t)N * HD;     // [P, 2]

  const size_t NH = (size_t)N * HD;
  float* P0  = (float*)d_ws;      // projection (ppi)  [N,128]
  float* P1  = P0 + NH;           // projection (sim)  [N,128]
  float* AG  = P1 + NH;           // shared accumulator [N,128]
  float* IOp = AG + NH;           // inv_out_ppi [N]
  float* IIp = IOp + N;           // inv_in_ppi  [N]
  float* IOs = IIp + N;           // inv_out_sim [N]
  float* IIs = IOs + N;           // inv_in_sim  [N]

  const int ntiles = (N + 31) >> 5;
  const int gGemm  = ntiles < 2048 ? ntiles : 2048;
  const int gEdge1 = grid_cap((long)E1, 8, 8192);   // 8 waves/block
  const int gEdge2 = grid_cap((long)E2, 8, 8192);
  const int gElem  = grid_cap((long)(NH / 4), 256, 4096);
  const int gDeg1  = grid_cap((long)E1, 256, 4096);
  const int gDeg2  = grid_cap((long)E2, 256, 4096);
  const int gNode  = grid_cap((long)4 * N, 256, 2048);
  const int gPair  = grid_cap((long)P, 8, 8192);

  // ---- degrees -> 1/sqrt(max(deg,1)) (layer-invariant, computed once)
  zero_f32<<<gNode, 256, 0, stream>>>(IOp, (long)4 * N);
  degree_accum<<<gDeg1, 256, 0, stream>>>(src_ppi, dst_ppi, IOp, IIp, E1);
  degree_accum<<<gDeg2, 256, 0, stream>>>(src_sim, dst_sim, IOs, IIs, E2);
  inv_sqrt_deg<<<gNode, 256, 0, stream>>>(IOp, 4 * N);

  // ---- layer 1
  gemm_n128<<<gGemm, 128, 0, stream>>>(feat, W0_ppi, P0, N);
  gemm_n128<<<gGemm, 128, 0, stream>>>(feat, W0_sim, P1, N);
  zero_f32<<<gElem, 256, 0, stream>>>(AG, (long)NH);
  scatter_edges<<<gEdge1, 256, 0, stream>>>(P0, src_ppi, dst_ppi, IOp, IIp, AG, E1);
  scatter_edges<<<gEdge2, 256, 0, stream>>>(P1, src_sim, dst_sim, IOs, IIs, AG, E2);
  combine_relu<<<gElem, 256, 0, stream>>>(AG, b0_ppi, b0_sim, zout, (long)(NH / 4));

  // ---- layer 2 (reads h1 from zout, overwrites zout with final z)
  gemm_n128<<<gGemm, 128, 0, stream>>>(zout, W1_ppi, P0, N);
  gemm_n128<<<gGemm, 128, 0, stream>>>(zout, W1_sim, P1, N);
  zero_f32<<<gElem, 256, 0, stream>>>(AG, (long)NH);
  scatter_edges<<<gEdge1, 256, 0, stream>>>(P0, src_ppi, dst_ppi, IOp, IIp, AG, E1);
  scatter_edges<<<gEdge2, 256, 0, stream>>>(P1, src_sim, dst_sim, IOs, IIs, AG, E2);
  combine_relu<<<gElem, 256, 0, stream>>>(AG, b1_ppi, b1_sim, zout, (long)(NH / 4));

  // ---- pair classifier
  classifier<<<gPair, 256, 0, stream>>>(zout, pair_src, pair_dst, Wc, bc, logits, P);
}